// NonlinearStringForce_68118181314857
// MI455X (gfx1250) — compile-verified
//
#include <hip/hip_runtime.h>

// out[b,m] = -sum_{i,j,k} A[m,i,j,k] q[b,i] q[b,j] q[b,k]
// == -( A2[4096,4096] @ P^T[4096,512] ) contracted with q over i.
// A2[(m*64+i),(j*64+k)] is A's native flat layout; P[b,(j*64+k)] = q[b,j]*q[b,k].

typedef float v2f __attribute__((ext_vector_type(2)));
typedef float v8f __attribute__((ext_vector_type(8)));

#define NMODES 64
#define NBATCH 512
#define KDIM   4096   // 64*64
#define KT     64     // K tile
#define LSTR   68     // LDS row stride in floats (68*4=272B: 16B aligned rows, no bank conflicts)
#define BT     64     // batch tile per workgroup

// ---------------- prologue: P[b, j*64+k] = q[b,j]*q[b,k] ----------------
__global__ __launch_bounds__(256) void nsf_outer_kernel(const float* __restrict__ q,
                                                        float* __restrict__ P) {
  int idx = blockIdx.x * 256 + threadIdx.x;      // 0 .. 512*4096-1
  int b  = idx >> 12;
  int jk = idx & 4095;
  P[idx] = q[b * 64 + (jk >> 6)] * q[b * 64 + (jk & 63)];
}

// ---------------- main: fused WMMA GEMM + i-contraction ----------------
__global__ __launch_bounds__(128) void nsf_wmma_kernel(const float* __restrict__ A,
                                                       const float* __restrict__ P,
                                                       const float* __restrict__ q,
                                                       float* __restrict__ out) {
  __shared__ float Abuf[2][64 * LSTR];
  __shared__ float Pbuf[2][64 * LSTR];
  __shared__ float spart[4][BT];

  const int tid  = threadIdx.x;
  const int lane = tid & 31;
  const int wid  = tid >> 5;                 // wave id: i-strip = wid*16
  const int m    = blockIdx.x >> 3;          // 0..63
  const int b0   = (blockIdx.x & 7) * BT;    // 0..448

  const float* Arows = A + (size_t)m  * 64 * KDIM;  // 64 rows (i) x 4096 (jk)
  const float* Prows = P + (size_t)b0 * KDIM;       // 64 rows (b) x 4096 (jk)

  v8f acc[4] = {};

  // Issue one 64x64 tile of A and of P into LDS buffer `buf` via async copies.
  auto issue_tile = [&](int buf, int kk) {
#pragma unroll
    for (int c = 0; c < 8; ++c) {
      int id  = c * 128 + tid;        // 0..1023
      int row = id >> 4;              // 0..63
      int col = (id & 15) << 2;       // 0..60 (16B chunks)
      unsigned la = (unsigned)(unsigned long long)(&Abuf[buf][row * LSTR + col]);
      unsigned long long ga = (unsigned long long)(Arows + (size_t)row * KDIM + kk + col);
      asm volatile("global_load_async_to_lds_b128 %0, %1, off"
                   :: "v"(la), "v"(ga) : "memory");
      unsigned lp = (unsigned)(unsigned long long)(&Pbuf[buf][row * LSTR + col]);
      unsigned long long gp = (unsigned long long)(Prows + (size_t)row * KDIM + kk + col);
      asm volatile("global_load_async_to_lds_b128 %0, %1, off"
                   :: "v"(lp), "v"(gp) : "memory");
    }
  };

  // Consume one K-chunk from LDS buffer `buf`.
  const int il = lane & 15;          // row-within-16 (i for A-frag, b for B-frag)
  const int kh = (lane >> 4) << 1;   // lanes 16-31 hold k+2,k+3
  auto compute_tile = [&](int buf) {
    const float* Ab = &Abuf[buf][0];
    const float* Pb = &Pbuf[buf][0];
#pragma unroll 4
    for (int k4 = 0; k4 < KT; k4 += 4) {
      v2f afr = *(const v2f*)&Ab[(wid * 16 + il) * LSTR + k4 + kh];
#pragma unroll
      for (int t = 0; t < 4; ++t) {
        v2f bfr = *(const v2f*)&Pb[(t * 16 + il) * LSTR + k4 + kh];
        acc[t] = __builtin_amdgcn_wmma_f32_16x16x4_f32(
            /*neg_a=*/false, afr, /*neg_b=*/false, bfr,
            /*c_mod=*/(short)0, acc[t], /*reuse_a=*/false, /*reuse_b=*/false);
      }
    }
  };

  issue_tile(0, 0);
  for (int kt = 0; kt < KDIM / KT; ++kt) {
    if (kt + 1 < KDIM / KT) {
      issue_tile((kt + 1) & 1, (kt + 1) * KT);
      asm volatile("s_wait_asynccnt 16" ::: "memory");  // oldest 16 (current tile) done
    } else {
      asm volatile("s_wait_asynccnt 0" ::: "memory");
    }
    __syncthreads();
    compute_tile(kt & 1);
    __syncthreads();
  }

  // Epilogue: out[b,m] = -sum_i q[b,i] * G[i,b]
  // C/D layout: VGPR r -> rows (r, r+8); lanes j and j+16 hold the same column b.
  const int half = lane >> 4;
  float psum[4];
#pragma unroll
  for (int t = 0; t < 4; ++t) {
    int b = b0 + t * 16 + il;
    float s = 0.f;
#pragma unroll
    for (int r = 0; r < 8; ++r) {
      int ig = wid * 16 + half * 8 + r;
      s += q[b * 64 + ig] * acc[t][r];
    }
    s += __shfl_xor(s, 16, 32);      // fold the two row-halves (same b)
    psum[t] = s;
  }
  if (half == 0) {
#pragma unroll
    for (int t = 0; t < 4; ++t) spart[wid][t * 16 + il] = psum[t];
  }
  __syncthreads();
  if (tid < BT) {
    float r = spart[0][tid] + spart[1][tid] + spart[2][tid] + spart[3][tid];
    out[(size_t)(b0 + tid) * 64 + m] = -r;
  }
}

extern "C" void kernel_launch(void* const* d_in, const int* in_sizes, int n_in,
                              void* d_out, int out_size, void* d_ws, size_t ws_size,
                              hipStream_t stream) {
  const float* q = (const float*)d_in[0];   // [512, 64]
  const float* A = (const float*)d_in[1];   // [64,64,64,64] == [4096, 4096]
  float* P   = (float*)d_ws;                // [512, 4096] scratch (8 MB)
  float* out = (float*)d_out;               // [512, 64]

  nsf_outer_kernel<<<(NBATCH * KDIM) / 256, 256, 0, stream>>>(q, P);
  nsf_wmma_kernel<<<NMODES * (NBATCH / BT), 128, 0, stream>>>(A, P, q, out);
}